// BoundaryAwareLoss_34754875359832
// MI455X (gfx1250) — compile-verified
//
#include <hip/hip_runtime.h>
#include <hip/hip_bf16.h>
#include <math.h>

// Boundary-aware masked BCE loss, MI455X (gfx1250, wave32).
// Bandwidth-bound: ~144 MiB HBM traffic -> ~6.2 us floor at 23.3 TB/s.
// CDNA5 features: global_load_async_to_lds_b128 (ASYNCcnt), s_wait_asynccnt,
// wave32 ballots for bit-packed binary erosion. WMMA intentionally unused:
// the workload has no matmul structure (elementwise + 3D stencil + reduce).
//
// Per block: one (b, ci) channel, a 32(x) x 16(y) x 8(z) tile.
// Targets halo staged to LDS as 40-float rows starting at x0-4 so every
// 16B chunk is 16B-aligned and fully in- or out-of-range in x -> exact
// per-lane predication of b128 async copies. Rows are ballot-packed into
// 40-bit masks; the 2-iteration 6-neighbor erosion (L1 ball radius 2)
// becomes ~10 shift/AND ops per row on uint64.

#define TXX 32
#define TYY 16
#define TZZ 8
#define HALO_X 40              // 4 + TXX + 4, 16B-aligned row of 10 x b128
#define HALO_Y 20              // TYY + 4
#define HALO_Z 12              // TZZ + 4
#define NROWS (HALO_Z * HALO_Y)        // 240 rows
#define TTILE_F (NROWS * HALO_X)       // 9600 floats = 38.4 KB
#define E1_Z 10
#define E1_Y 18
#define NBLOCKS 4096
#define BOUNDARY_W_MINUS1 4.0f

static __device__ __forceinline__ unsigned lds_low32(const void* p) {
  // Flat LDS addresses carry the wave-relative LDS byte offset in the low
  // 32 bits (aperture mapping).
  return (unsigned)(unsigned long long)p;
}

static __device__ __forceinline__ void async_g2l_b128(unsigned lds_byte,
                                                      const float* sbase,
                                                      unsigned byte_off) {
  // GVS form: vdst = LDS byte offset, vaddr = 32-bit byte offset, saddr = base.
  asm volatile("global_load_async_to_lds_b128 %0, %1, %2"
               :: "v"(lds_byte), "v"(byte_off), "s"(sbase)
               : "memory");
}

static __device__ __forceinline__ void wait_async0() {
#if defined(__has_builtin) && __has_builtin(__builtin_amdgcn_s_wait_asynccnt)
  __builtin_amdgcn_s_wait_asynccnt(0);
#else
  asm volatile("s_wait_asynccnt 0" ::: "memory");
#endif
}

// Cross erosion along x on a bit-row: needs m(x-1), m(x), m(x+1).
static __device__ __forceinline__ unsigned long long cx(unsigned long long m) {
  return m & (m << 1) & (m >> 1);
}

__global__ __launch_bounds__(256)
void ba_loss_main(const float* __restrict__ logits,
                  const float* __restrict__ targets,
                  const unsigned char* __restrict__ mask8,  // numpy bool (1B)
                  const float* __restrict__ spatial,
                  float* __restrict__ partials) {
  const int tid  = threadIdx.x;
  const int lane = tid & 31;
  const int wv   = tid >> 5;           // 8 waves
  const unsigned bid = blockIdx.x;

  // Decode: [b:1][ci:2][tileZ:4][tileY:3][tileX:2]
  const int tileX = bid & 3;
  const int tileY = (bid >> 2) & 7;
  const int tileZ = (bid >> 5) & 15;
  const int ci    = (bid >> 9) & 3;
  const int b     = (bid >> 11) & 1;

  const int x0 = tileX << 5;
  const int y0 = tileY << 4;
  const int z0 = tileZ << 3;
  const int c  = 2 * ci + 1;           // INSTANCE_INDICES = (1,3,5,7)

  // Early-exit: channel masked off -> contributes 0 to num and den.
  const unsigned char m = mask8[b * 8 + c];
  if (m == 0) {
    if (tid == 0) { partials[2 * bid] = 0.0f; partials[2 * bid + 1] = 0.0f; }
    return;
  }

  const size_t volOff = ((size_t)(b * 8 + c)) << 21;   // * 128^3
  const float* __restrict__ logC = logits  + volOff;
  const float* __restrict__ tgtC = targets + volOff;
  const float* __restrict__ spB  = spatial + (((size_t)b) << 21);

  __shared__ __align__(16) float ttile[TTILE_F];
  __shared__ unsigned long long Mrow[NROWS];           // 40-bit row masks
  __shared__ unsigned long long E1s[E1_Z * E1_Y];      // erosion pass 1
  __shared__ unsigned long long E2s[TZZ * TYY];        // erosion pass 2
  __shared__ float wsum_n[8], wsum_d[8];

  // ---- Phase 1: zero the halo tile (covers zero-padding at volume borders)
  {
    float4 z4 = make_float4(0.f, 0.f, 0.f, 0.f);
    float4* t4 = (float4*)ttile;
    for (int i = tid; i < TTILE_F / 4; i += 256) t4[i] = z4;
  }
  __syncthreads();

  // ---- Phase 2: async global -> LDS b128 copy of targets halo rows.
  // 3 rows per wave issue: lanes 0..29 cover 3 rows x 10 chunks of 16B.
  {
    const unsigned ttBase = lds_low32(ttile);
    const int sub = lane / 10;               // 0..2 (3 = idle lanes 30,31)
    const int li  = lane - sub * 10;         // 16B chunk within row, 0..9
    for (int r = wv; r < NROWS / 3; r += 8) {        // 80 triples
      const int row = r * 3 + sub;
      const int tz  = row / HALO_Y;
      const int ty  = row - tz * HALO_Y;
      const int gz  = z0 - 2 + tz;
      const int gy  = y0 - 2 + ty;
      const int gxb = x0 - 4 + li * 4;       // first of 4 floats, 16B-aligned
      if (sub < 3 && (unsigned)gz < 128u && (unsigned)gy < 128u &&
          (unsigned)gxb < 128u) {
        const unsigned elem = ((unsigned)gz * 128u + (unsigned)gy) * 128u +
                              (unsigned)gxb;
        async_g2l_b128(ttBase + (unsigned)(row * HALO_X + li * 4) * 4u,
                       tgtC, elem * 4u);
      }
    }
  }
  wait_async0();
  __syncthreads();

  // ---- Phase 3: ballot floats -> 40-bit row masks (wave32)
  for (int r = wv; r < NROWS; r += 8) {
    float v0 = ttile[r * HALO_X + lane];
    int p0 = v0 > 0.5f;
    int p1 = 0;
    if (lane < 8) {
      float v1 = ttile[r * HALO_X + 32 + lane];
      p1 = v1 > 0.5f;
    }
    unsigned long long b0 = __ballot(p0);
    unsigned long long b1 = __ballot(p1);
    if (lane == 0)
      Mrow[r] = (b0 & 0xffffffffull) | (b1 << 32);
  }
  __syncthreads();

  // ---- Phase 4: erosion pass 1 (bits 3..36 valid; bit i <-> gx = x0-4+i)
  if (tid < E1_Z * E1_Y) {
    const int z1 = tid / E1_Y;
    const int y1 = tid - z1 * E1_Y;
    unsigned long long cen = Mrow[(z1 + 1) * HALO_Y + (y1 + 1)];
    unsigned long long e = cx(cen)
        & Mrow[(z1    ) * HALO_Y + (y1 + 1)]
        & Mrow[(z1 + 2) * HALO_Y + (y1 + 1)]
        & Mrow[(z1 + 1) * HALO_Y + (y1    )]
        & Mrow[(z1 + 1) * HALO_Y + (y1 + 2)];
    E1s[tid] = e;
  }
  __syncthreads();

  // ---- Phase 5: erosion pass 2 (bits 4..35 valid = the 32 tile columns)
  if (tid < TZZ * TYY) {
    const int z2 = tid / TYY;
    const int y2 = tid - z2 * TYY;
    unsigned long long cen = E1s[(z2 + 1) * E1_Y + (y2 + 1)];
    unsigned long long e = cx(cen)
        & E1s[(z2    ) * E1_Y + (y2 + 1)]
        & E1s[(z2 + 2) * E1_Y + (y2 + 1)]
        & E1s[(z2 + 1) * E1_Y + (y2    )]
        & E1s[(z2 + 1) * E1_Y + (y2 + 2)];
    E2s[tid] = e;
  }
  __syncthreads();

  // ---- Phase 6: fused BCE + boundary weight + masks, accumulate
  float num = 0.0f, den = 0.0f;
  const int tx = lane;
  const int bitpos = tx + 4;
  for (int z = 0; z < TZZ; ++z) {
    for (int h = 0; h < 2; ++h) {
      const int y = wv + (h << 3);     // wv in [0,8), y in [0,16)
      const unsigned long long Mr = Mrow[(z + 2) * HALO_Y + (y + 2)];
      const unsigned long long Er = E2s[z * TYY + y];
      const unsigned tbit = (unsigned)(Mr >> bitpos) & 1u;
      const unsigned ebit = (unsigned)(Er >> bitpos) & 1u;
      const float tf  = (float)tbit;
      const float wgt = 1.0f + BOUNDARY_W_MINUS1 * (float)(tbit & (ebit ^ 1u));
      const unsigned idx =
          (((unsigned)(z0 + z) * 128u + (unsigned)(y0 + y)) * 128u) +
          (unsigned)(x0 + tx);
      const float l  = logC[idx];
      const float sm = spB[idx];
      const float bce = fmaxf(l, 0.0f) - l * tf + log1pf(expf(-fabsf(l)));
      num = fmaf(wgt * sm, bce, num);
      den += sm;
    }
  }

  // ---- Phase 7: wave32 shuffle reduce, then 8 wave leaders via LDS
  for (int off = 16; off > 0; off >>= 1) {
    num += __shfl_down(num, off, 32);
    den += __shfl_down(den, off, 32);
  }
  if (lane == 0) { wsum_n[wv] = num; wsum_d[wv] = den; }
  __syncthreads();
  if (tid == 0) {
    float N = 0.0f, D = 0.0f;
    for (int i = 0; i < 8; ++i) { N += wsum_n[i]; D += wsum_d[i]; }
    partials[2 * bid]     = N;
    partials[2 * bid + 1] = D;
  }
}

__global__ __launch_bounds__(256)
void ba_loss_reduce(const float* __restrict__ partials,
                    float* __restrict__ out, int nblocks) {
  float n = 0.0f, d = 0.0f;
  for (int i = threadIdx.x; i < nblocks; i += 256) {
    n += partials[2 * i];
    d += partials[2 * i + 1];
  }
  for (int off = 16; off > 0; off >>= 1) {
    n += __shfl_down(n, off, 32);
    d += __shfl_down(d, off, 32);
  }
  __shared__ float sn[8], sd[8];
  const int lane = threadIdx.x & 31;
  const int wv   = threadIdx.x >> 5;
  if (lane == 0) { sn[wv] = n; sd[wv] = d; }
  __syncthreads();
  if (threadIdx.x == 0) {
    float N = 0.0f, D = 0.0f;
    for (int i = 0; i < 8; ++i) { N += sn[i]; D += sd[i]; }
    out[0] = (D > 0.0f) ? (N / fmaxf(D, 1.0f)) : 0.0f;
  }
}

extern "C" void kernel_launch(void* const* d_in, const int* in_sizes, int n_in,
                              void* d_out, int out_size, void* d_ws, size_t ws_size,
                              hipStream_t stream) {
  (void)in_sizes; (void)n_in; (void)out_size; (void)ws_size;
  const float* logits         = (const float*)d_in[0];
  const float* targets        = (const float*)d_in[1];
  const unsigned char* mask8  = (const unsigned char*)d_in[2]; // jax bool = 1B
  const float* spatial        = (const float*)d_in[3];
  float* partials = (float*)d_ws;        // NBLOCKS * 2 floats = 32 KB
  float* out      = (float*)d_out;

  ba_loss_main<<<dim3(NBLOCKS), dim3(256), 0, stream>>>(
      logits, targets, mask8, spatial, partials);
  ba_loss_reduce<<<dim3(1), dim3(256), 0, stream>>>(partials, out, NBLOCKS);
}